// GAT_54185307406459
// MI455X (gfx1250) — compile-verified
//
#include <hip/hip_runtime.h>
#include <math.h>

// Problem constants (from reference setup_inputs)
#define BB 16
#define NN 512
#define TT 12
#define DD 128
#define FF 128
#define BTT (BB * TT)          // 192 fused batch*time slices
#define NEG_INF_C (-1.0e9f)
#define SLOPE 0.2f

// LDS row pitch for the staged x tile: 128 + 4 pad DWORDs -> stride 132.
// (132 mod 64 == 4, so the 16 M-lanes of a half map to 16 distinct bank
//  pairs for ds_load_b64 -> conflict-free fragment reads.)
#define XPITCH 132

typedef float v2f __attribute__((ext_vector_type(2)));
typedef float v8f __attribute__((ext_vector_type(8)));
typedef float v4f __attribute__((ext_vector_type(4)));
typedef unsigned int u32x4 __attribute__((ext_vector_type(4)));
typedef int i32x4 __attribute__((ext_vector_type(4)));
typedef int i32x8 __attribute__((ext_vector_type(8)));

#if __has_builtin(__builtin_amdgcn_tensor_load_to_lds)
#define USE_TDM 1
#else
#define USE_TDM 0
#endif

__device__ __forceinline__ float lane_shfl(float v, int src) {
    return __shfl(v, src, 32);
}
__device__ __forceinline__ float lane_shfl_xor(float v, int mask) {
    return __shfl_xor(v, mask, 32);
}

// ---------------------------------------------------------------------------
// Kernel A: h[bt,n,f] = x[b,n,t,:] @ W[:,f]  via V_WMMA_F32_16X16X4_F32.
// The 16x128 x tile (shared by all 8 waves) is staged into LDS once per WG by
// the Tensor Data Mover (single DMA descriptor, padded rows), removing the 8x
// duplicated global reads. Also emits e_src = h.a_src, e_dst = h.a_dst.
// ---------------------------------------------------------------------------
__global__ __launch_bounds__(256)
void gat_proj_kernel(const float* __restrict__ x, const float* __restrict__ W,
                     const float* __restrict__ a_src, const float* __restrict__ a_dst,
                     float* __restrict__ h_ws, float* __restrict__ esrc_ws,
                     float* __restrict__ edst_ws) {
    const int bx   = blockIdx.x;
    const int bt   = bx >> 5;          // 0..191
    const int tile = bx & 31;          // n-tile index
    const int b    = bt / TT;
    const int t    = bt % TT;
    const int n0   = tile * 16;

    const int tid  = threadIdx.x;
    const int wave = tid >> 5;         // 0..7
    const int lane = tid & 31;
    const int half = lane >> 4;        // K-half select for 16x16x4 fragments
    const int lm   = lane & 15;        // M (A/C row) or N (B col) index
    const int f0   = wave * 16;

    __shared__ float lds_x[16 * XPITCH];   // staged x tile, row pitch 132
    __shared__ float lds_s[8][16];
    __shared__ float lds_d[8][16];

    // Base of the x tile: x[b, n0, t, 0]; row stride T*D elements.
    const float* xbase = x + ((size_t)(b * NN + n0) * TT + t) * DD;

#if USE_TDM
    if (wave == 0) {
        // ---- Tensor DMA descriptor (D#), per CDNA5 ISA ch.8.3/8.4 ----
        const unsigned long long gaddr = (unsigned long long)(uintptr_t)xbase;
        const unsigned int lds_off = (unsigned int)(uintptr_t)(&lds_x[0]);

        u32x4 g0;
        g0[0] = 1u;                                   // count=1 valid descriptor
        g0[1] = lds_off;                              // lds_addr (bytes)
        g0[2] = (unsigned int)(gaddr & 0xffffffffu);  // global_addr[31:0]
        g0[3] = (unsigned int)((gaddr >> 32) & 0x01ffffffu) | (2u << 30); // addr[56:32] | type=2

        // data_size=2 (4B), pad_enable, pad_interval code 6 (=128 DWORDs),
        // pad_amount code 3 (=4 DWORDs)  -> LDS row pitch 132 DWORDs
        const unsigned int d0 = (2u << 16) | (1u << 20) | (6u << 22) | (3u << 25);
        const unsigned int tdim0 = DD;            // tensor row length (elems)
        const unsigned int tdim1 = NN;            // rows available (OOB bound)
        const unsigned int stride0 = TT * DD;     // 1536 elems between rows
        i32x8 g1;
        g1[0] = (int)d0;
        g1[1] = (int)((tdim0 & 0xffffu) << 16);                    // dim0[15:0]
        g1[2] = (int)((tdim0 >> 16) | ((tdim1 & 0xffffu) << 16));  // dim0[31:16] | dim1[15:0]
        g1[3] = (int)((tdim1 >> 16) | (128u << 16));               // dim1[31:16] | tile_dim0=128
        g1[4] = (int)(16u);                                        // tile_dim1=16, tile_dim2=0
        g1[5] = (int)stride0;                                      // dim0_stride[31:0]
        g1[6] = 0;                                                 // dim0_stride[47:32] | dim1_stride[15:0]
        g1[7] = 0;                                                 // dim1_stride[47:16] (unused, 2D tile)

        i32x4 gz4 = {0, 0, 0, 0};
        i32x8 gz8 = {0, 0, 0, 0, 0, 0, 0, 0};
        // 6-arg form (amdgpu-toolchain / clang-23): groups 2,3 unused (2D),
        // extra group zero-filled, cpol=0.
        __builtin_amdgcn_tensor_load_to_lds(g0, g1, gz4, gz4, gz8, 0);
        __builtin_amdgcn_s_wait_tensorcnt(0);
    }
    __syncthreads();
#else
    // Fallback: cooperative staging (2 x b128 per thread), still deduplicated.
    {
        const int flat = tid * 8;               // 256 threads * 8 floats = 8KB
        const int row  = flat >> 7;             // /128
        const int col  = flat & 127;
        const float* src = xbase + (size_t)row * (TT * DD) + col;
        v4f v0 = *(const v4f*)(src);
        v4f v1 = *(const v4f*)(src + 4);
        float* dst = &lds_x[row * XPITCH + col];
        *(v4f*)(dst)     = v0;
        *(v4f*)(dst + 4) = v1;
    }
    __syncthreads();
#endif

    // B-matrix column for this lane: W[:, f0+lm], stride FF
    const float* wcol = W + (size_t)(f0 + lm);
    const float* xl   = &lds_x[lm * XPITCH];    // this lane's A row in LDS

    v8f c = {};
#pragma unroll
    for (int k0 = 0; k0 < DD; k0 += 4) {
        const int d0 = k0 + 2 * half;                 // this lane's K pair
        v2f a = *(const v2f*)(xl + d0);               // ds_load_b64, bank-clean
        v2f bf;
        bf.x = wcol[(size_t)d0 * FF];                 // B[d0,   lm+f0]
        bf.y = wcol[(size_t)(d0 + 1) * FF];           // B[d0+1, lm+f0]
        c = __builtin_amdgcn_wmma_f32_16x16x4_f32(false, a, false, bf,
                                                  (short)0, c, false, false);
    }

    // Store h tile (C layout: VGPR r -> row r + 8*half, col f0+lm) and build
    // per-lane partials of e_src/e_dst.
    const float asf = a_src[f0 + lm];
    const float adf = a_dst[f0 + lm];
    float ps[8], pd[8];
#pragma unroll
    for (int r = 0; r < 8; ++r) {
        const int n = n0 + r + 8 * half;
        h_ws[(size_t)(bt * NN + n) * FF + f0 + lm] = c[r];
        ps[r] = c[r] * asf;
        pd[r] = c[r] * adf;
    }
    // Reduce over the 16 f-lanes of each half (xor masks 1,2,4,8 stay in-half)
#pragma unroll
    for (int m = 1; m < 16; m <<= 1) {
#pragma unroll
        for (int r = 0; r < 8; ++r) {
            ps[r] += lane_shfl_xor(ps[r], m);
            pd[r] += lane_shfl_xor(pd[r], m);
        }
    }
    if (lm < 8) {                                     // lanes 0-7 and 16-23
        const int row = (lane & 7) + 8 * half;        // rows 0-7 / 8-15
        lds_s[wave][row] = ps[lane & 7];
        lds_d[wave][row] = pd[lane & 7];
    }
    __syncthreads();
    if (tid < 16) {
        float s = 0.f;
#pragma unroll
        for (int w = 0; w < 8; ++w) s += lds_s[w][tid];
        esrc_ws[bt * NN + n0 + tid] = s;
    } else if (tid < 32) {
        const int r = tid - 16;
        float s = 0.f;
#pragma unroll
        for (int w = 0; w < 8; ++w) s += lds_d[w][r];
        edst_ws[bt * NN + n0 + r] = s;
    }
}

// ---------------------------------------------------------------------------
// Kernel B: fused mask + online-softmax + (alpha @ h) + ELU.
// One WG per (bt, 16-row i-tile); 8 waves each own a 16-wide f-block.
// alpha is never materialized (would be 201 MB). h reads hit L2 (50 MB < 192 MB).
// ---------------------------------------------------------------------------
__global__ __launch_bounds__(256)
void gat_attn_kernel(const float* __restrict__ h_ws, const float* __restrict__ esrc_ws,
                     const float* __restrict__ edst_ws, const int* __restrict__ adj,
                     float* __restrict__ out) {
    const int bx   = blockIdx.x;
    const int bt   = bx >> 5;
    const int tile = bx & 31;
    const int b    = bt / TT;
    const int t    = bt % TT;
    const int i0   = tile * 16;

    const int tid  = threadIdx.x;
    const int wave = tid >> 5;
    const int lane = tid & 31;
    const int half = lane >> 4;
    const int lm   = lane & 15;
    const int f0   = wave * 16;

    const int   irow   = i0 + lm;                 // this lane's attention row
    const float es_i   = esrc_ws[bt * NN + irow];
    const int*  adjrow = adj + (size_t)irow * NN;

    float m_run = -3.0e38f;   // running row max (per lane-row lm, both halves)
    float l_run = 0.0f;       // running row sum
    v8f   acc   = {};

    for (int jt = 0; jt < NN / 16; ++jt) {
        const int j0 = jt * 16;
        // e_dst for the whole j-tile, one value per lane, redistributed by shfl
        const float ed_lane = edst_ws[bt * NN + j0 + lm];

        float ev[8];
        float tmax = -3.0e38f;
#pragma unroll
        for (int u = 0; u < 8; ++u) {
            const int q  = u >> 1, s = u & 1;
            const int kl = 4 * q + 2 * half + s;  // local K index (A layout)
            float e = es_i + lane_shfl(ed_lane, kl);
            e = (e > 0.f) ? e : SLOPE * e;        // LeakyReLU
            const int j = j0 + kl;
            const bool valid = (adjrow[j] != 0) || (j == irow);
            e = valid ? e : NEG_INF_C;
            ev[u] = e;
            tmax = fmaxf(tmax, e);
        }
        tmax = fmaxf(tmax, lane_shfl_xor(tmax, 16));  // combine K-halves
        const float m_new = fmaxf(m_run, tmax);

        float p[8];
        float tsum = 0.f;
#pragma unroll
        for (int u = 0; u < 8; ++u) {
            p[u] = expf(ev[u] - m_new);           // masked -> exp(-1e9) -> 0
            tsum += p[u];
        }
        tsum += lane_shfl_xor(tsum, 16);
        const float scale = expf(m_run - m_new);
        l_run = l_run * scale + tsum;
        m_run = m_new;

        // Rescale accumulator: C-layout row of VGPR r is (r + 8*half)
#pragma unroll
        for (int r = 0; r < 8; ++r)
            acc[r] *= lane_shfl(scale, r + 8 * half);

        // acc += P(16x16) @ Hj(16x16-block of F) via four 16x16x4 WMMAs
        const float* hb = h_ws + (size_t)(bt * NN + j0) * FF + f0 + lm;
#pragma unroll
        for (int q = 0; q < 4; ++q) {
            const int kk = 4 * q + 2 * half;
            v2f af; af.x = p[2 * q]; af.y = p[2 * q + 1];
            v2f bf; bf.x = hb[(size_t)kk * FF]; bf.y = hb[(size_t)(kk + 1) * FF];
            acc = __builtin_amdgcn_wmma_f32_16x16x4_f32(false, af, false, bf,
                                                        (short)0, acc, false, false);
        }
    }

    // Normalize by row sum, apply ELU, write out[b, n, t, f]
    const float linv = 1.0f / l_run;
#pragma unroll
    for (int r = 0; r < 8; ++r) {
        float v = acc[r] * lane_shfl(linv, r + 8 * half);
        v = (v > 0.f) ? v : (expf(v) - 1.0f);     // ELU (alpha=1)
        const int n = i0 + r + 8 * half;
        out[((size_t)(b * NN + n) * TT + t) * FF + f0 + lm] = v;
    }
}

// ---------------------------------------------------------------------------
extern "C" void kernel_launch(void* const* d_in, const int* in_sizes, int n_in,
                              void* d_out, int out_size, void* d_ws, size_t ws_size,
                              hipStream_t stream) {
    const float* x     = (const float*)d_in[0];
    const float* W     = (const float*)d_in[1];
    const float* a_src = (const float*)d_in[2];
    const float* a_dst = (const float*)d_in[3];
    const int*   adj   = (const int*)d_in[4];
    float*       outp  = (float*)d_out;

    // Workspace: h [192*512*128] f32 (50.3 MB), then e_src, e_dst [192*512] each
    float* h_ws = (float*)d_ws;
    float* esrc = h_ws + (size_t)BTT * NN * FF;
    float* edst = esrc + (size_t)BTT * NN;

    const dim3 grid(BTT * (NN / 16));   // 6144 workgroups
    const dim3 block(256);              // 8 wave32

    gat_proj_kernel<<<grid, block, 0, stream>>>(x, W, a_src, a_dst, h_ws, esrc, edst);
    gat_attn_kernel<<<grid, block, 0, stream>>>(h_ws, esrc, edst, adj, outp);

    (void)in_sizes; (void)n_in; (void)out_size; (void)ws_size;
}